// VHACoding_10806137717261
// MI455X (gfx1250) — compile-verified
//
#include <hip/hip_runtime.h>

typedef __attribute__((ext_vector_type(2))) float v2f;
typedef __attribute__((ext_vector_type(4))) float v4f;
typedef __attribute__((ext_vector_type(8))) float v8f;

#define BI 128
#define BC 128
#define NR 36
#define NW 50
#define EMB 1024
#define MASKV (-1.0f)

#define CPW 4                 // captions per workgroup
#define MPAD 48               // NR padded to 3 M-tiles of 16
#define NSLOT 64              // words per caption padded to 4 N-tiles
#define NTOT (CPW * NSLOT)    // 256 N-slots per workgroup
#define KC 32                 // K chunk in f32 elements
#define NCHUNK (EMB / KC)     // 32
#define LDA 36                // LDS row stride: 16B-aligned, bank-conflict-free
#define LDB 36
#define ASZ (MPAD * LDA)      // 1728 floats per buffer
#define BSZ (NTOT * LDB)      // 9216 floats per buffer

__global__ __launch_bounds__(256)
void vha_simmax_wmma(const float* __restrict__ imgs,
                     const float* __restrict__ caps,
                     const int* __restrict__ img_lens,
                     const int* __restrict__ cap_lens,
                     float* __restrict__ out)
{
  __shared__ __align__(16) float As[2 * ASZ];   // double-buffered A: 48 x 36 f32
  __shared__ __align__(16) float Bs[2 * BSZ];   // double-buffered B: 256 x 36 f32

  const int cTile = blockIdx.x;       // 0..31 (group of 4 captions)
  const int img   = blockIdx.y;       // 0..127
  const int tid   = threadIdx.x;
  const int lane  = tid & 31;
  const int wid   = tid >> 5;         // 0..7
  const int l16   = lane & 15;
  const int half  = lane >> 4;        // 0/1
  const int kb    = half * 2;         // K pair selected by lane half
  const int nbase = wid * 32;         // 2 N-tiles per wave

  const int ilen = img_lens[img];

  const float* imgBase = imgs + (size_t)img * NR * EMB;
  const float* capBase = caps + (size_t)cTile * CPW * NW * EMB;

  // ---- per-thread staging geometry (32-bit offsets -> saddr+voffset loads) ----
  const int  s4   = (tid & 7) * 4;          // f32 column within chunk
  const int  ar0  = tid >> 3;               // A pass 0: rows 0..31 (< NR always)
  const int  ar1  = 32 + (tid >> 3);        // A pass 1: rows 32..47 (tid < 128)
  const bool a1st = (tid < 128);
  const bool a1ld = a1st && (ar1 < NR);
  const int  aOff0 = ar0 * EMB + s4;
  const int  aOff1 = ar1 * EMB + s4;

  int bSlot[8], bOff[8];
  bool bVal[8];
  #pragma unroll
  for (int j = 0; j < 8; ++j) {
    bSlot[j] = (tid >> 3) + 32 * j;         // 0..255
    int w = bSlot[j] & 63, cl = bSlot[j] >> 6;
    bVal[j] = (w < NW);
    bOff[j] = (cl * NW + w) * EMB + s4;
  }

  // staging registers; invalid/padded entries stay zero forever
  const v4f vz4 = {0.f, 0.f, 0.f, 0.f};
  v4f ra0 = vz4, ra1 = vz4, rb[8];
  #pragma unroll
  for (int j = 0; j < 8; ++j) rb[j] = vz4;

  auto loadRegs = [&](int kc) {
    ra0 = *(const v4f*)(imgBase + aOff0 + kc);
    if (a1ld) ra1 = *(const v4f*)(imgBase + aOff1 + kc);
    #pragma unroll
    for (int j = 0; j < 8; ++j)
      if (bVal[j]) rb[j] = *(const v4f*)(capBase + bOff[j] + kc);
  };

  auto storeLDS = [&](int buf) {
    float* Ab = As + buf * ASZ;
    float* Bb = Bs + buf * BSZ;
    *(v4f*)(Ab + ar0 * LDA + s4) = ra0;
    if (a1st) *(v4f*)(Ab + ar1 * LDA + s4) = ra1;
    #pragma unroll
    for (int j = 0; j < 8; ++j)
      *(v4f*)(Bb + bSlot[j] * LDB + s4) = rb[j];
  };

  const v8f vz8 = {0.f,0.f,0.f,0.f,0.f,0.f,0.f,0.f};
  v8f acc[3][2];
  #pragma unroll
  for (int mT = 0; mT < 3; ++mT)
    #pragma unroll
    for (int nT = 0; nT < 2; ++nT)
      acc[mT][nT] = vz8;

  auto compute = [&](int buf) {
    const float* Ab = As + buf * ASZ;
    const float* Bb = Bs + buf * BSZ;
    #pragma unroll
    for (int kk = 0; kk < KC / 4; ++kk) {
      v2f a[3], b[2];
      #pragma unroll
      for (int mT = 0; mT < 3; ++mT)
        a[mT] = *(const v2f*)(Ab + (mT * 16 + l16) * LDA + kk * 4 + kb);
      #pragma unroll
      for (int nT = 0; nT < 2; ++nT)
        b[nT] = *(const v2f*)(Bb + (nbase + nT * 16 + l16) * LDB + kk * 4 + kb);
      #pragma unroll
      for (int mT = 0; mT < 3; ++mT)
        #pragma unroll
        for (int nT = 0; nT < 2; ++nT)
          acc[mT][nT] = __builtin_amdgcn_wmma_f32_16x16x4_f32(
              false, a[mT], false, b[nT], (short)0, acc[mT][nT], false, false);
    }
  };

  // ---- software pipeline: regs hold chunk c+1 while WMMA eats chunk c ----
  loadRegs(0);
  storeLDS(0);
  loadRegs(KC);                         // chunk 1 in flight
  __syncthreads();                      // buffer 0 ready

  for (int c = 0; c < NCHUNK; ++c) {
    const int p = c & 1;
    if (c + 1 < NCHUNK) storeLDS(1 - p);          // chunk c+1 -> other buffer
    if (c + 2 < NCHUNK) loadRegs((c + 2) * KC);   // chunk c+2 global loads in flight
    compute(p);                                   // 48 WMMAs hide the latency
    __syncthreads();                              // one barrier per chunk
  }

  // ---- fused epilogue: masked max over regions, write out[img, cap, word] ----
  const int rbase = half * 8;   // C/D layout: lane half selects rows {0..7} / {8..15}
  #pragma unroll
  for (int nT = 0; nT < 2; ++nT) {
    float cmax = -__builtin_huge_valf();
    #pragma unroll
    for (int mT = 0; mT < 3; ++mT) {
      #pragma unroll
      for (int v = 0; v < 8; ++v) {
        int row = mT * 16 + rbase + v;
        float s = acc[mT][nT][v];
        cmax = (row < ilen) ? fmaxf(cmax, s) : cmax;
      }
    }
    cmax = fmaxf(cmax, __shfl_xor(cmax, 16, 32));   // merge the two lane halves
    if (ilen < NR) cmax = fmaxf(cmax, MASKV);       // masked rows hold -1 in reference max
    int slot = nbase + nT * 16 + l16;
    int cl = slot >> 6, w = slot & 63;
    if (half == 0 && w < NW) {
      int cap = cTile * CPW + cl;
      float val = (w < cap_lens[cap]) ? cmax : MASKV;
      out[((size_t)img * BC + cap) * NW + w] = val;
    }
  }
}

extern "C" void kernel_launch(void* const* d_in, const int* in_sizes, int n_in,
                              void* d_out, int out_size, void* d_ws, size_t ws_size,
                              hipStream_t stream) {
  const float* imgs     = (const float*)d_in[0];
  const float* caps     = (const float*)d_in[1];
  const int*   img_lens = (const int*)d_in[2];
  const int*   cap_lens = (const int*)d_in[3];
  float*       out      = (float*)d_out;

  dim3 grid(BC / CPW, BI);   // (32, 128) workgroups
  vha_simmax_wmma<<<grid, 256, 0, stream>>>(imgs, caps, img_lens, cap_lens, out);
}